// CTCLoss_71743133713011
// MI455X (gfx1250) — compile-verified
//
#include <hip/hip_runtime.h>
#include <cstdint>

// CTC forward loss, MI455X (gfx1250).
// One workgroup per batch row (N=512 blocks). Sequential DP over T with:
//  - 8-deep async global->LDS DMA ring (ASYNCcnt) for emission rows,
//    issued 7 steps ahead, drained with s_wait_asynccnt 6 (in-order loads)
//  - LDS ping-pong alpha lattice, one wave32 split-barrier per step
//  - 160 threads (5 waves), 2 lattice slots per thread for trans-op ILP.

#define T_DIM 512
#define N_DIM 512
#define C_DIM 80
#define S_DIM 128
#define L_DIM 257          // 2*S + 1
#define NEGV  (-1.0e30f)
#define BLOCK 160          // 5 wave32 waves; 2 lattice slots per thread
#define SLOT_STRIDE 160
#define ROW_LANES 20       // 20 lanes * 16B(B128) = 320B = one 80-f32 row
#define RING 8             // emission ring depth (2.5KB LDS)

__device__ __forceinline__ void async_row_load_b128(uint32_t lds_dst,
                                                    unsigned long long gbase,
                                                    uint32_t voff) {
  // GVS mode: mem_addr = SADDR + VADDR ; VDST holds the LDS byte address.
  asm volatile("global_load_async_to_lds_b128 %0, %1, %2"
               :
               : "v"(lds_dst), "v"(voff), "s"(gbase)
               : "memory");
}

__device__ __forceinline__ void wait_async_le6() {
  asm volatile("s_wait_asynccnt 6" ::: "memory");
}

__global__ __launch_bounds__(BLOCK) void ctc_alpha_kernel(
    const float* __restrict__ log_probs,   // (T, N, C)
    const int*   __restrict__ targets,     // (N, S)
    const int*   __restrict__ input_len,   // (N,)
    const int*   __restrict__ target_len,  // (N,)
    float*       __restrict__ out)         // (N,)
{
  __shared__ __align__(16) float emitBuf[RING][C_DIM];  // emission ring
  __shared__ float alphaA[L_DIM];
  __shared__ float alphaB[L_DIM];

  const int n   = blockIdx.x;
  const int tid = threadIdx.x;

  const int inLen = input_len[n];
  const int tl    = target_len[n];

  // Per-thread lattice metadata for slots s = tid and s = tid+160.
  // ext[s]: blank(0) for even s, targets[s/2] for odd s.
  // skip_ok[s]: only odd s>=3 with targets[k] != targets[k-1].
  int  ext_s[2] = {0, 0};
  bool skip[2]  = {false, false};
#pragma unroll
  for (int j = 0; j < 2; ++j) {
    const int s = tid + j * SLOT_STRIDE;
    if (s < L_DIM && (s & 1)) {
      const int k = s >> 1;
      ext_s[j] = targets[n * S_DIM + k];
      if (s >= 3) skip[j] = (ext_s[j] != targets[n * S_DIM + k - 1]);
    }
  }

  const unsigned long long rowBase =
      (unsigned long long)(uintptr_t)(log_probs + (size_t)n * C_DIM);
  const unsigned long long rowStride =
      (unsigned long long)N_DIM * C_DIM * sizeof(float);

  const uint32_t voff = (uint32_t)(tid * 16);
  uint32_t ldsDst[RING];
#pragma unroll
  for (int r = 0; r < RING; ++r)
    ldsDst[r] = (uint32_t)(uintptr_t)(&emitBuf[r][0]) + voff;

  // Prologue: fill the ring with rows 0..7.  After wait<=6 (in-order async
  // completion), rows 0 and 1 are guaranteed resident.
  if (tid < ROW_LANES) {
#pragma unroll
    for (int r = 0; r < RING; ++r)
      async_row_load_b128(ldsDst[r],
                          rowBase + (unsigned long long)r * rowStride, voff);
  }
  wait_async_le6();
  __syncthreads();

  // alpha(t=0): only slots 0 (blank) and 1 (first label) reachable.
#pragma unroll
  for (int j = 0; j < 2; ++j) {
    const int s = tid + j * SLOT_STRIDE;
    if (s < L_DIM) {
      float a = NEGV;
      if (s == 0)      a = emitBuf[0][0];
      else if (s == 1) a = emitBuf[0][ext_s[j]];
      alphaA[s] = a;
    }
  }
  __syncthreads();

  float* pc = alphaA;   // alpha(t-1)
  float* pn = alphaB;   // alpha(t)

  for (int t = 1; t < T_DIM; ++t) {
    const int buf = t & (RING - 1);

    // Issue DMA of row t+7 into slot (t+7)&7 == (t-1)&7: all reads of that
    // slot happened before the barrier we just passed.  Completion is only
    // required 7 steps from now (wait<=6 below).
    const int tp = t + RING - 1;
    if (tp < T_DIM && tid < ROW_LANES) {
      async_row_load_b128(ldsDst[tp & (RING - 1)],
                          rowBase + (unsigned long long)tp * rowStride, voff);
    }

    const bool live = (t < inLen);
#pragma unroll
    for (int j = 0; j < 2; ++j) {
      const int s = tid + j * SLOT_STRIDE;
      if (s < L_DIM) {
        const float a0 = pc[s];
        const float a1 = (s >= 1) ? pc[s - 1] : NEGV;
        const float a2 = skip[j] ? pc[s - 2] : NEGV;
        float m = fmaxf(a0, fmaxf(a1, a2));
        const float lse =
            m + __logf(__expf(a0 - m) + __expf(a1 - m) + __expf(a2 - m));
        const float em = emitBuf[buf][ext_s[j]];
        pn[s] = live ? (lse + em) : a0;   // freeze once t >= input_length
      }
    }

    wait_async_le6();     // in-order asyncs: guarantees row t+1 is resident
    __syncthreads();      // publishes pn, protects pc + ring slot reuse

    float* tmp = pc; pc = pn; pn = tmp;
  }

  // Final: logsumexp of slots 2*tl (last blank) and 2*tl-1 (last label).
  if (tid == 0) {
    const int i1 = 2 * tl;
    const float v1 = pc[i1];
    const float v2 = pc[i1 - 1];
    const float m  = fmaxf(v1, v2);
    float loss = -(m + __logf(__expf(v1 - m) + __expf(v2 - m)));
    // zero_infinity: NaN / inf / >=1e10 -> 0   (NaN fails both comparisons)
    if (!(loss < 1.0e10f) || !(loss > -1.0e10f)) loss = 0.0f;
    out[n] = loss;
  }
}

extern "C" void kernel_launch(void* const* d_in, const int* in_sizes, int n_in,
                              void* d_out, int out_size, void* d_ws, size_t ws_size,
                              hipStream_t stream) {
  (void)in_sizes; (void)n_in; (void)out_size; (void)d_ws; (void)ws_size;
  const float* log_probs  = (const float*)d_in[0];   // (T, N, C) f32
  const int*   targets    = (const int*)d_in[1];     // (N, S)    i32
  const int*   input_len  = (const int*)d_in[2];     // (N,)      i32
  const int*   target_len = (const int*)d_in[3];     // (N,)      i32
  float*       out        = (float*)d_out;           // (N,)      f32

  ctc_alpha_kernel<<<dim3(N_DIM), dim3(BLOCK), 0, stream>>>(
      log_probs, targets, input_len, target_len, out);
}